// CompGCNBase_24721831755947
// MI455X (gfx1250) — compile-verified
//
#include <hip/hip_runtime.h>
#include <math.h>

// ---------------------------------------------------------------------------
// CompGCN (2 layers) for MI455X / gfx1250, wave32, WMMA f32 16x16x4.
//
// ccorr(a, rel) @ W  ==  a @ Q   with Q[i][j] = sum_k rel[(i+k)%256] * W[k][j]
// Edges counting-sorted by relation; one 512-thread WG per relation builds
// Q_r in LDS (256KB of the 320KB WGP LDS) with WMMA, holds a 256x16 Q strip
// per wave in 128 VGPRs, and streams double-buffered 16-edge tiles via
// GLOBAL_LOAD_ASYNC_TO_LDS_B128 (ASYNCcnt) into v_wmma_f32_16x16x4_f32,
// scattering results with global_atomic_add_f32.
// ---------------------------------------------------------------------------

#define N_NODES   50000
#define DIM       256
#define NREL      474          // edge relation types (2*NUM_RELS)
#define NE_HALF   300000
#define NBASES    50
#define BN_EPS    1e-5f
#define XS_STRIDE 260          // floats; 1040B row pitch: 16B-aligned, bank-spread

typedef float v2f __attribute__((ext_vector_type(2)));
typedef float v8f __attribute__((ext_vector_type(8)));
typedef int   v4i __attribute__((vector_size(16)));   // matches builtin param

#define AS_GLOBAL __attribute__((address_space(1)))
#define AS_LDS    __attribute__((address_space(3)))

__device__ __forceinline__ v8f wmma4(v2f a, v2f b, v8f c) {
  // (neg_a, A, neg_b, B, c_mod, C, reuse_a, reuse_b)
  return __builtin_amdgcn_wmma_f32_16x16x4_f32(false, a, false, b, (short)0, c,
                                               false, false);
}

// --- gfx1250 async global->LDS copy (ASYNCcnt tracked, bypasses VGPRs) -----
__device__ __forceinline__ void async_load_b128(const float* g, float* l) {
  float* gm = const_cast<float*>(g);
#if __has_builtin(__builtin_amdgcn_global_load_async_to_lds_b128)
  __builtin_amdgcn_global_load_async_to_lds_b128(
      (AS_GLOBAL v4i*)(AS_GLOBAL float*)gm,
      (AS_LDS v4i*)(AS_LDS float*)l, 0, 0);
#else
  asm volatile("global_load_async_to_lds_b128 %0, %1, off"
               :
               : "v"((unsigned)(unsigned long long)(AS_LDS float*)l),
                 "v"((unsigned long long)gm)
               : "memory");
#endif
}

__device__ __forceinline__ void wait_async0() {
#if __has_builtin(__builtin_amdgcn_s_wait_asynccnt)
  __builtin_amdgcn_s_wait_asynccnt(0);
#else
  asm volatile("s_wait_asynccnt 0x0" ::: "memory");
#endif
}

__device__ __forceinline__ void wait_async2() {
#if __has_builtin(__builtin_amdgcn_s_wait_asynccnt)
  __builtin_amdgcn_s_wait_asynccnt(2);
#else
  asm volatile("s_wait_asynccnt 0x2" ::: "memory");
#endif
}

// ---------------------------------------------------------------- utilities
__global__ void zero_kernel(float* p, int n) {
  int i = blockIdx.x * 256 + threadIdx.x;
  if (i < n) p[i] = 0.0f;
}

__global__ void copy256_kernel(const float* __restrict__ s, float* __restrict__ d) {
  d[threadIdx.x] = s[threadIdx.x];
}

__global__ void hist_deg_kernel(const int* __restrict__ ei, const int* __restrict__ et,
                                int* cntIn, int* cntOut, float* degIn, float* degOut) {
  int e = blockIdx.x * 256 + threadIdx.x;
  if (e >= 2 * NE_HALF) return;
  if (e < NE_HALF) {
    atomicAdd(&cntIn[et[e]], 1);
    atomicAdd(&degIn[ei[e]], 1.0f);
  } else {
    atomicAdd(&cntOut[et[e]], 1);
    atomicAdd(&degOut[ei[e]], 1.0f);
  }
}

__global__ void scan_kernel(const int* __restrict__ cntIn, const int* __restrict__ cntOut,
                            int* offIn, int* offOut) {
  if (threadIdx.x == 0) {
    int s = 0;
    for (int i = 0; i < NREL; ++i) { offIn[i] = s; s += cntIn[i]; }
    offIn[NREL] = s;
  }
  if (threadIdx.x == 1) {
    int s = 0;
    for (int i = 0; i < NREL; ++i) { offOut[i] = s; s += cntOut[i]; }
    offOut[NREL] = s;
  }
}

__global__ void scatter_kernel(const int* __restrict__ et,
                               const int* __restrict__ offIn, const int* __restrict__ offOut,
                               int* curIn, int* curOut, int* sortIn, int* sortOut) {
  int e = blockIdx.x * 256 + threadIdx.x;
  if (e >= 2 * NE_HALF) return;
  if (e < NE_HALF) {
    int t = et[e];
    sortIn[offIn[t] + atomicAdd(&curIn[t], 1)] = e;
  } else {
    int l = e - NE_HALF, t = et[e];
    sortOut[offOut[t] + atomicAdd(&curOut[t], 1)] = l;
  }
}

__global__ void norm_kernel(const int* __restrict__ ei,
                            const float* __restrict__ degIn, const float* __restrict__ degOut,
                            float* nrmIn, float* nrmOut) {
  int e = blockIdx.x * 256 + threadIdx.x;
  if (e >= 2 * NE_HALF) return;
  int s = ei[e];
  int d = ei[2 * NE_HALF + e];
  if (e < NE_HALF) {
    float a = degIn[s], b = degIn[d];
    float ia = a > 0.0f ? 1.0f / sqrtf(a) : 0.0f;
    float ib = b > 0.0f ? 1.0f / sqrtf(b) : 0.0f;
    nrmIn[e] = ia * ib;
  } else {
    float a = degOut[s], b = degOut[d];
    float ia = a > 0.0f ? 1.0f / sqrtf(a) : 0.0f;
    float ib = b > 0.0f ? 1.0f / sqrtf(b) : 0.0f;
    nrmOut[e - NE_HALF] = ia * ib;
  }
}

// rel_tab[0..473] = rel_wt @ rel_basis ; rel_tab[474] = loop_rel
__global__ void rel0_kernel(const float* __restrict__ rel_wt,
                            const float* __restrict__ rel_basis,
                            const float* __restrict__ loopRel,
                            float* __restrict__ relTab) {
  int idx = blockIdx.x * 256 + threadIdx.x;
  if (idx >= (NREL + 1) * DIM) return;
  int r = idx >> 8, c = idx & 255;
  if (r < NREL) {
    float s = 0.0f;
    for (int b = 0; b < NBASES; ++b)
      s += rel_wt[r * NBASES + b] * rel_basis[b * DIM + c];
    relTab[idx] = s;
  } else {
    relTab[idx] = loopRel[c];
  }
}

// Q[i][j] = sum_k rv[(i+k)%256] * W[k][j]  (single relation vector) -> global
__global__ __launch_bounds__(512) void qgen_kernel(const float* __restrict__ rvG,
                                                   const float* __restrict__ W,
                                                   float* __restrict__ Q) {
  __shared__ float rv2[520];   // rel | rel | rel[0:8] -> no modulo in inner loop
  int tid = threadIdx.x;
  if (tid < 256) {
    float v = rvG[tid];
    rv2[tid] = v;
    rv2[tid + 256] = v;
  }
  if (tid < 8) rv2[512 + tid] = rvG[tid];
  __syncthreads();
  int wave = tid >> 5, lane = tid & 31;
  int nA = lane & 15, kb = (lane < 16) ? 0 : 2;
  int ntBase = wave * 16, hi = (lane < 16) ? 0 : 8;
  int mt = blockIdx.x;
  int i = mt * 16 + nA;
  v8f c = {};
#pragma unroll
  for (int kk = 0; kk < 64; ++kk) {
    int k0 = kk * 4 + kb;
    v2f a, b;
    a.x = rv2[i + k0];
    a.y = rv2[i + k0 + 1];
    b.x = W[k0 * 256 + ntBase + nA];
    b.y = W[(k0 + 1) * 256 + ntBase + nA];
    c = wmma4(a, b, c);
  }
#pragma unroll
  for (int v = 0; v < 8; ++v)
    Q[(mt * 16 + v + hi) * 256 + ntBase + nA] = c[v];
}

// Generic C[M x 256] = A[M x 256] @ B[256 x 256] (store). 8 row-tiles per WG,
// B-strip register-resident (64 x v2f = 128 VGPRs/lane), b128 staging.
__global__ __launch_bounds__(512) void gemm_kernel(const float* __restrict__ A,
                                                   const float* __restrict__ B,
                                                   float* __restrict__ C, int M) {
  __shared__ __align__(16) float xs[16 * XS_STRIDE];
  int tid = threadIdx.x, wave = tid >> 5, lane = tid & 31;
  int nA = lane & 15, kb = (lane < 16) ? 0 : 2;
  int ntBase = wave * 16, hi = (lane < 16) ? 0 : 8;
  v2f Bf[64];
#pragma unroll
  for (int kk = 0; kk < 64; ++kk) {
    int k0 = kk * 4 + kb;
    Bf[kk].x = B[k0 * 256 + ntBase + nA];
    Bf[kk].y = B[(k0 + 1) * 256 + ntBase + nA];
  }
  for (int t = 0; t < 8; ++t) {
    int rowBase = blockIdx.x * 128 + t * 16;
    if (rowBase >= M) break;                    // uniform across the block
    __syncthreads();
#pragma unroll
    for (int j = 0; j < 2; ++j) {
      int chunk = tid + j * 512;                // 1024 x 16B chunks
      int row = chunk >> 6;
      int cf = (chunk & 63) << 2;               // float offset in row
      int r = rowBase + row;
      if (r >= M) r = M - 1;
      *(float4*)(xs + row * XS_STRIDE + cf) =
          *(const float4*)(A + (size_t)r * 256 + cf);
    }
    __syncthreads();
    v8f c = {};
#pragma unroll
    for (int kk = 0; kk < 64; ++kk) {
      int k0 = kk * 4 + kb;
      v2f a = *(const v2f*)(xs + nA * XS_STRIDE + k0);
      c = wmma4(a, Bf[kk], c);
    }
#pragma unroll
    for (int v = 0; v < 8; ++v) {
      int row = rowBase + v + hi;
      if (row < M) C[row * 256 + ntBase + nA] = c[v];
    }
  }
}

// ---------------------------------------------------------------- edge kernel
// LDS floats: Q 65536 | rv2 520 | xs0 4160 | xs1 4160 | meta 3*48
#define EDGE_SMEM_FLOATS (65536 + 520 + 2 * 16 * XS_STRIDE + 144)
#define EDGE_SMEM_BYTES  (EDGE_SMEM_FLOATS * 4)

__global__ __launch_bounds__(512) void edge_kernel(
    const float* __restrict__ X, const float* __restrict__ relTab,
    const float* __restrict__ W, const int* __restrict__ srcArr,
    const int* __restrict__ dstArr, const int* __restrict__ sortIdx,
    const int* __restrict__ binOff, const float* __restrict__ nrm,
    float* __restrict__ accum) {
  extern __shared__ float smem[];
  float* Qs  = smem;                        // 256x256 Q_r (row-major)
  float* rv2 = Qs + 65536;                  // 520: rel|rel|rel[0:8]
  float* xs0 = rv2 + 520;                   // tile buffer 0 (16 x XS_STRIDE)
  float* xs1 = xs0 + 16 * XS_STRIDE;        // tile buffer 1
  int*   msrc = (int*)(xs1 + 16 * XS_STRIDE);  // 3 x 16 (triple-buffered meta)
  int*   mdst = msrc + 48;                     // 3 x 16
  float* mnrm = (float*)(mdst + 48);           // 3 x 16

  const int r = blockIdx.x;
  const int e0 = binOff[r], e1 = binOff[r + 1];
  if (e0 >= e1) return;

  const int tid = threadIdx.x;
  const int wave = tid >> 5, lane = tid & 31;
  const int nA = lane & 15, kb = (lane < 16) ? 0 : 2;
  const int ntBase = wave * 16, hi = (lane < 16) ? 0 : 8;

  if (tid < 256) {
    float v = relTab[r * 256 + tid];
    rv2[tid] = v;
    rv2[tid + 256] = v;
  }
  if (tid < 8) rv2[512 + tid] = relTab[r * 256 + tid];
  __syncthreads();

  // Phase 1: Q_r = S(rel_r) @ W into LDS. Wave owns N-tile `wave`, loops M-tiles.
  {
    v2f Wf[64];
#pragma unroll
    for (int kk = 0; kk < 64; ++kk) {
      int k0 = kk * 4 + kb;
      Wf[kk].x = W[k0 * 256 + ntBase + nA];
      Wf[kk].y = W[(k0 + 1) * 256 + ntBase + nA];
    }
    for (int mt = 0; mt < 16; ++mt) {
      int i = mt * 16 + nA;
      v8f c = {};
#pragma unroll
      for (int kk = 0; kk < 64; ++kk) {
        int k0 = kk * 4 + kb;
        v2f a;
        a.x = rv2[i + k0];
        a.y = rv2[i + k0 + 1];
        c = wmma4(a, Wf[kk], c);
      }
#pragma unroll
      for (int v = 0; v < 8; ++v)
        Qs[(mt * 16 + v + hi) * 256 + ntBase + nA] = c[v];
    }
  }
  __syncthreads();

  // Phase 2: per-wave register-resident 256x16 strip of Q (128 VGPRs).
  v2f Qf[64];
#pragma unroll
  for (int kk = 0; kk < 64; ++kk) {
    int k0 = kk * 4 + kb;
    Qf[kk].x = Qs[k0 * 256 + ntBase + nA];
    Qf[kk].y = Qs[(k0 + 1) * 256 + ntBase + nA];
  }

  // Phase 3: double-buffered 16-edge tiles with async global->LDS staging.
  auto writeMeta = [&](int slot, int b) {
    if (tid < 16) {
      int t = b + tid;
      int s = 0, d = 0;
      float nn = 0.0f;
      if (t < e1) {
        int e = sortIdx[t];
        s = srcArr[e];
        d = dstArr[e];
        nn = nrm[e];
      }
      msrc[slot * 16 + tid] = s;
      mdst[slot * 16 + tid] = d;
      mnrm[slot * 16 + tid] = nn;
    }
  };
  auto issueStage = [&](float* buf, int slot) {
#pragma unroll
    for (int j = 0; j < 2; ++j) {
      int chunk = tid + j * 512;                 // 1024 x 16B chunks / tile
      int row = chunk >> 6;
      int cf = (chunk & 63) << 2;
      async_load_b128(X + (size_t)msrc[slot * 16 + row] * 256 + cf,
                      buf + row * XS_STRIDE + cf);
    }
  };

  writeMeta(0, e0);
  __syncthreads();
  issueStage(xs0, 0);

  int tIdx = 0;
  for (int base = e0; base < e1; base += 16, ++tIdx) {
    const int curSlot = tIdx % 3;
    const float* xsC = (tIdx & 1) ? xs1 : xs0;
    const bool hasNext = (base + 16) < e1;       // uniform across block
    if (hasNext) writeMeta((tIdx + 1) % 3, base + 16);
    __syncthreads();                             // meta visible; prev reads done
    if (hasNext) {
      issueStage((tIdx & 1) ? xs0 : xs1, (tIdx + 1) % 3);
      wait_async2();                             // current tile's copies done
    } else {
      wait_async0();
    }
    __syncthreads();                             // staged data visible to all

    v8f c = {};
#pragma unroll
    for (int kk = 0; kk < 64; ++kk) {
      int k0 = kk * 4 + kb;
      v2f a = *(const v2f*)(xsC + nA * XS_STRIDE + k0);
      c = wmma4(a, Qf[kk], c);
    }
#pragma unroll
    for (int v = 0; v < 8; ++v) {
      int row = v + hi;
      float val = c[v] * mnrm[curSlot * 16 + row];
      atomicAdd(&accum[mdst[curSlot * 16 + row] * 256 + ntBase + nA], val);
    }
  }
}

// ---------------------------------------------------------------- epilogues
__global__ void tanh3_kernel(float* a, int n) {
  int i = blockIdx.x * 256 + threadIdx.x;
  if (i < n) a[i] = tanhf(a[i] * (1.0f / 3.0f));
}

__global__ void bn_stats_kernel(const float* __restrict__ a, float* s1, float* s2) {
  int c = threadIdx.x;
  int r0 = blockIdx.x * 256;
  int rEnd = r0 + 256;
  if (rEnd > N_NODES) rEnd = N_NODES;
  float s = 0.0f, ss = 0.0f;
  for (int r = r0; r < rEnd; ++r) {
    float v = a[r * 256 + c] * (1.0f / 3.0f);
    s += v;
    ss += v * v;
  }
  atomicAdd(&s1[c], s);
  atomicAdd(&s2[c], ss);
}

__global__ void bn_apply_kernel(float* a, const float* __restrict__ s1,
                                const float* __restrict__ s2) {
  int i = blockIdx.x * 256 + threadIdx.x;
  if (i >= N_NODES * 256) return;
  int c = i & 255;
  float mu = s1[c] * (1.0f / (float)N_NODES);
  float var = s2[c] * (1.0f / (float)N_NODES) - mu * mu;
  if (var < 0.0f) var = 0.0f;
  float inv = 1.0f / sqrtf(var + BN_EPS);
  a[i] = tanhf((a[i] * (1.0f / 3.0f) - mu) * inv);
}

// ---------------------------------------------------------------- launch
extern "C" void kernel_launch(void* const* d_in, const int* in_sizes, int n_in,
                              void* d_out, int out_size, void* d_ws, size_t ws_size,
                              hipStream_t stream) {
  (void)in_sizes; (void)n_in; (void)out_size; (void)ws_size;

  const float* x         = (const float*)d_in[0];
  const float* rel_basis = (const float*)d_in[1];
  const float* rel_wt    = (const float*)d_in[2];
  const float* loop_rel1 = (const float*)d_in[3];
  const float* w_in1     = (const float*)d_in[4];
  const float* w_out1    = (const float*)d_in[5];
  const float* w_loop1   = (const float*)d_in[6];
  const float* w_rel1    = (const float*)d_in[7];
  const float* loop_rel2 = (const float*)d_in[8];
  const float* w_in2     = (const float*)d_in[9];
  const float* w_out2    = (const float*)d_in[10];
  const float* w_loop2   = (const float*)d_in[11];
  const float* w_rel2    = (const float*)d_in[12];
  const int*   ei        = (const int*)d_in[13];
  const int*   et        = (const int*)d_in[14];

  float* x2 = (float*)d_out;                 // 50000*256
  float* r2 = x2 + N_NODES * DIM;            // 474*256

  // ---- workspace carve (~58 MB total)
  char* base = (char*)d_ws;
  size_t off = 0;
  auto alloc = [&](size_t bytes) -> char* {
    char* p = base + off;
    off += (bytes + 255) & ~(size_t)255;
    return p;
  };
  float* bufA  = (float*)alloc((size_t)N_NODES * DIM * 4);  // layer1 accum -> x1
  float* Qloop = (float*)alloc((size_t)DIM * DIM * 4);
  float* relT1 = (float*)alloc((size_t)(NREL + 1) * DIM * 4);
  float* relT2 = (float*)alloc((size_t)(NREL + 1) * DIM * 4);
  char* zbase = base + off;                                  // zero-init block
  float* degIn  = (float*)alloc((size_t)N_NODES * 4);
  float* degOut = (float*)alloc((size_t)N_NODES * 4);
  int* cntIn  = (int*)alloc(480 * 4);
  int* cntOut = (int*)alloc(480 * 4);
  int* curIn  = (int*)alloc(480 * 4);
  int* curOut = (int*)alloc(480 * 4);
  float* bnS = (float*)alloc(256 * 4);
  float* bnQ = (float*)alloc(256 * 4);
  size_t zfloats = (size_t)((base + off) - zbase) / 4;
  int* offIn  = (int*)alloc(480 * 4);
  int* offOut = (int*)alloc(480 * 4);
  float* nrmIn  = (float*)alloc((size_t)NE_HALF * 4);
  float* nrmOut = (float*)alloc((size_t)NE_HALF * 4);
  int* sortIn  = (int*)alloc((size_t)NE_HALF * 4);
  int* sortOut = (int*)alloc((size_t)NE_HALF * 4);

  (void)hipFuncSetAttribute(reinterpret_cast<const void*>(&edge_kernel),
                            hipFuncAttributeMaxDynamicSharedMemorySize,
                            EDGE_SMEM_BYTES);

  const int gEdges = (2 * NE_HALF + 255) / 256;
  const int gNodesElems = (N_NODES * DIM + 255) / 256;
  const int gGemmNodes = (N_NODES + 127) / 128;   // 391
  const int gGemmRel = (NREL + 127) / 128;        // 4

  // ---- setup
  zero_kernel<<<(int)((zfloats + 255) / 256), 256, 0, stream>>>((float*)zbase, (int)zfloats);
  hist_deg_kernel<<<gEdges, 256, 0, stream>>>(ei, et, cntIn, cntOut, degIn, degOut);
  scan_kernel<<<1, 32, 0, stream>>>(cntIn, cntOut, offIn, offOut);
  scatter_kernel<<<gEdges, 256, 0, stream>>>(et, offIn, offOut, curIn, curOut, sortIn, sortOut);
  norm_kernel<<<gEdges, 256, 0, stream>>>(ei, degIn, degOut, nrmIn, nrmOut);
  rel0_kernel<<<((NREL + 1) * DIM + 255) / 256, 256, 0, stream>>>(rel_wt, rel_basis, loop_rel1, relT1);

  // ---- layer 1
  qgen_kernel<<<16, 512, 0, stream>>>(relT1 + NREL * DIM, w_loop1, Qloop);
  gemm_kernel<<<gGemmNodes, 512, 0, stream>>>(x, Qloop, bufA, N_NODES);   // loop term (store)
  edge_kernel<<<NREL, 512, EDGE_SMEM_BYTES, stream>>>(
      x, relT1, w_in1, ei, ei + 2 * NE_HALF, sortIn, offIn, nrmIn, bufA);
  edge_kernel<<<NREL, 512, EDGE_SMEM_BYTES, stream>>>(
      x, relT1, w_out1, ei + NE_HALF, ei + 3 * NE_HALF, sortOut, offOut, nrmOut, bufA);
  tanh3_kernel<<<gNodesElems, 256, 0, stream>>>(bufA, N_NODES * DIM);     // x1
  gemm_kernel<<<gGemmRel, 512, 0, stream>>>(relT1, w_rel1, relT2, NREL);  // r1
  copy256_kernel<<<1, 256, 0, stream>>>(loop_rel2, relT2 + NREL * DIM);

  // ---- layer 2 (accumulate directly into d_out x2 region)
  qgen_kernel<<<16, 512, 0, stream>>>(relT2 + NREL * DIM, w_loop2, Qloop);
  gemm_kernel<<<gGemmNodes, 512, 0, stream>>>(bufA, Qloop, x2, N_NODES);
  edge_kernel<<<NREL, 512, EDGE_SMEM_BYTES, stream>>>(
      bufA, relT2, w_in2, ei, ei + 2 * NE_HALF, sortIn, offIn, nrmIn, x2);
  edge_kernel<<<NREL, 512, EDGE_SMEM_BYTES, stream>>>(
      bufA, relT2, w_out2, ei + NE_HALF, ei + 3 * NE_HALF, sortOut, offOut, nrmOut, x2);
  bn_stats_kernel<<<(N_NODES + 255) / 256, 256, 0, stream>>>(x2, bnS, bnQ);
  bn_apply_kernel<<<gNodesElems, 256, 0, stream>>>(x2, bnS, bnQ);
  gemm_kernel<<<gGemmRel, 512, 0, stream>>>(relT2, w_rel2, r2, NREL);     // r2
}